// OpticalDepth_55276229099579
// MI455X (gfx1250) — compile-verified
//
#include <hip/hip_runtime.h>

typedef __attribute__((ext_vector_type(2))) float v2f;
typedef __attribute__((ext_vector_type(8))) float v8f;

__constant__ int GAS_TAB[76] = {
  0,0,0,0,0,0,0,0,0,0,0,0,0,0,0,0,0,0,0,0,0,0,0,0,0,0,0,0,0,
  1,1,1,1,1,1,1,1,1,1,1,1,1,
  2,2,2,2,2,2,2,2,2,
  3,3,3,
  4,4,4,4,4,4,4,4,4,
  5,5,5,5,5,5,5,5,5,5,5,5,5
};
__constant__ int CHAN_TAB[76] = {
  0,1,2,3,4,5,6,7,8,9,10,11,12,13,14,15,16,17,18,19,20,21,22,23,24,25,26,27,28,
  0,1,2,17,18,19,20,21,22,25,26,27,28,
  0,1,2,5,6,9,10,13,14,
  0,1,2,
  0,1,2,3,4,7,8,11,12,
  0,1,2,15,16,17,18,19,20,21,22,27,28
};

// shared-memory layout (float indices)
#define W1_OFF   0      // 608
#define B1_OFF   608    // 304
#define W2_OFF   912    // 1216
#define B2_OFF   2128   // 304
#define W3_OFF   2432   // 304
#define B3_OFF   2736   // 76 (pad to 2816)
#define WLW_OFF  2816   // 29
#define BLW_OFF  2848   // 29
#define WIW_OFF  2880   // 29
#define BIW_OFF  2912   // 29 (pad to 2944)
#define GASI_OFF 2944   // 76 ints (pad to 3024)
#define CHNI_OFF 3024   // 76 ints (pad to 3104)
#define COMP_OFF 3104   // 128 floats per wave * 4 waves = 512
#define ACC_OFF  3616   // 512 floats per wave * 4 waves = 2048
#define SMEM_FLOATS 5664

__device__ __forceinline__ float elu1(float x) {
  // x>0 ? x : exp(x)-1  (branchless)
  return fmaxf(x, 0.0f) + (__expf(fminf(x, 0.0f)) - 1.0f);
}
__device__ __forceinline__ float sp1(float x) {
  // softplus, numerically stable
  return fmaxf(x, 0.0f) + __logf(1.0f + __expf(-fabsf(x)));
}

__global__ __launch_bounds__(128)
void OpticalDepth_55276229099579_kernel(
    const float* __restrict__ t_p, const float* __restrict__ comp,
    const float* __restrict__ null_lw, const float* __restrict__ null_iw,
    const float* __restrict__ W1, const float* __restrict__ b1,
    const float* __restrict__ W2, const float* __restrict__ b2,
    const float* __restrict__ W3, const float* __restrict__ b3,
    const float* __restrict__ W_lw, const float* __restrict__ b_lw,
    const float* __restrict__ W_iw, const float* __restrict__ b_iw,
    float* __restrict__ out, int Btot, int tilesTotal)
{
  __shared__ float smem[SMEM_FLOATS];
  const int tid = threadIdx.x, nt = blockDim.x;

  // ---- Phase 0: stage all weights/tables into LDS (once per block) ----
  for (int i = tid; i < 608;  i += nt) smem[W1_OFF + i] = W1[i];
  for (int i = tid; i < 304;  i += nt) smem[B1_OFF + i] = b1[i];
  for (int i = tid; i < 1216; i += nt) smem[W2_OFF + i] = W2[i];
  for (int i = tid; i < 304;  i += nt) smem[B2_OFF + i] = b2[i];
  for (int i = tid; i < 304;  i += nt) smem[W3_OFF + i] = W3[i];
  for (int i = tid; i < 76;   i += nt) smem[B3_OFF + i] = b3[i];
  for (int i = tid; i < 29;   i += nt) {
    smem[WLW_OFF + i] = W_lw[i]; smem[BLW_OFF + i] = b_lw[i];
    smem[WIW_OFF + i] = W_iw[i]; smem[BIW_OFF + i] = b_iw[i];
  }
  int* smi = (int*)smem;
  for (int i = tid; i < 76; i += nt) { smi[GASI_OFF + i] = GAS_TAB[i]; smi[CHNI_OFF + i] = CHAN_TAB[i]; }
  __syncthreads();

  const int lane = tid & 31;
  const int wave = tid >> 5;
  const int h    = lane >> 4;        // half-wave (splits K for A/B, M+8 for C/D)
  const bool hi  = (h != 0);
  const int m    = lane & 15;        // sample-in-tile (N) == unit row (M) for A
  const int ln   = m >> 2;           // local net 0..3
  const int j    = m & 3;            // hidden index within net
  const bool l0 = (ln == 0), l1 = (ln == 1), l2 = (ln == 2), l3 = (ln == 3);

  float* compW = smem + COMP_OFF + wave * 128;
  float* accW  = smem + ACC_OFF  + wave * 512;
  float* outLw = out + (size_t)Btot * 29;
  float* outIw = out + (size_t)Btot * 58;

  const int wavesTotal = gridDim.x * (nt >> 5);
  const int myWave = blockIdx.x * (nt >> 5) + wave;

  for (int t = myWave; t < tilesTotal; t += wavesTotal) {
    const int s0 = t << 4;

    // stage composition rows for the 16 samples of this tile (128 floats)
    float4 cc = *(const float4*)(comp + (size_t)s0 * 8 + lane * 4);
    *(float4*)(compW + lane * 4) = cc;

    // zero the 16x32 channel accumulator tile
    float4 z4 = make_float4(0.f, 0.f, 0.f, 0.f);
    *(float4*)(accW + lane * 16 + 0)  = z4;
    *(float4*)(accW + lane * 16 + 4)  = z4;
    *(float4*)(accW + lane * 16 + 8)  = z4;
    *(float4*)(accW + lane * 16 + 12) = z4;

    // layer-1 B operand: B(K,N)=t_p[N,K] for K<2, rows 2,3 = 0 (upper half lanes)
    float2 tp = ((const float2*)t_p)[s0 + m];
    v2f Btp = { hi ? 0.0f : tp.x, hi ? 0.0f : tp.y };

    for (int g = 0; g < 19; ++g) {
      const int n16 = g << 4;
      const int hb  = h << 3;  // 8h

      // layer-1 A: A(M=4ln+j, K=i)=W1[net,i,j]; K=2,3 padded with zero
      float w1a = smem[W1_OFF + (((g << 2) + ln) << 3) + j];
      float w1b = smem[W1_OFF + (((g << 2) + ln) << 3) + 4 + j];
      v2f A1 = { hi ? 0.0f : w1a, hi ? 0.0f : w1b };

      // layer-2 A chunk values: W2[net=4g+ln, i=v+2h, o=j]
      float w2v0 = smem[W2_OFF + (((g << 2) + ln) << 4) + (h << 3) + j];
      float w2v1 = smem[W2_OFF + (((g << 2) + ln) << 4) + (h << 3) + 4 + j];

      // biases: C/D vgpr r holds unit r+8h -> b*flat[16g + r + 8h]
      float4 biaA = *(const float4*)(smem + B1_OFF + n16 + hb);
      float4 biaB = *(const float4*)(smem + B1_OFF + n16 + hb + 4);
      float4 bibA = *(const float4*)(smem + B2_OFF + n16 + hb);
      float4 bibB = *(const float4*)(smem + B2_OFF + n16 + hb + 4);
      // layer-3 weights: nets (4g+2h) and (4g+1+2h)
      float4 w02 = *(const float4*)(smem + W3_OFF + n16 + hb);
      float4 w13 = *(const float4*)(smem + W3_OFF + n16 + hb + 4);
      const int nA = (g << 2) + (h << 1);
      const int nB = nA + 1;
      float b3A = smem[B3_OFF + nA];
      float b3B = smem[B3_OFF + nB];
      int gasA = smi[GASI_OFF + nA], gasB = smi[GASI_OFF + nB];
      int chA  = smi[CHNI_OFF + nA], chB  = smi[CHNI_OFF + nB];

      v8f zero8 = {0.f, 0.f, 0.f, 0.f, 0.f, 0.f, 0.f, 0.f};

      // ---- layer 1: one f32 WMMA per 4-net group ----
      v8f d1 = __builtin_amdgcn_wmma_f32_16x16x4_f32(false, A1, false, Btp,
                                                     (short)0, zero8, false, false);
      float h1v[8];
      h1v[0] = elu1(d1[0] + biaA.x); h1v[1] = elu1(d1[1] + biaA.y);
      h1v[2] = elu1(d1[2] + biaA.z); h1v[3] = elu1(d1[3] + biaA.w);
      h1v[4] = elu1(d1[4] + biaB.x); h1v[5] = elu1(d1[5] + biaB.y);
      h1v[6] = elu1(d1[6] + biaB.z); h1v[7] = elu1(d1[7] + biaB.w);

      // ---- C/D -> B layout junction via SWAPX16 swizzle (src = lane ^ 16) ----
      float sw[8];
      #pragma unroll
      for (int r = 0; r < 8; ++r)
        sw[r] = __int_as_float(__builtin_amdgcn_ds_swizzle(__float_as_int(h1v[r]), 0x401F));
      v2f B0  = { hi ? sw[2]  : h1v[0], hi ? sw[3]  : h1v[1] };
      v2f B1c = { hi ? sw[6]  : h1v[4], hi ? sw[7]  : h1v[5] };
      v2f B2c = { hi ? h1v[2] : sw[0],  hi ? h1v[3] : sw[1]  };
      v2f B3c = { hi ? h1v[6] : sw[4],  hi ? h1v[7] : sw[5]  };

      // block-diagonal A chunks (nonzero only for this lane's net)
      v2f A20 = { l0 ? w2v0 : 0.f, l0 ? w2v1 : 0.f };
      v2f A21 = { l1 ? w2v0 : 0.f, l1 ? w2v1 : 0.f };
      v2f A22 = { l2 ? w2v0 : 0.f, l2 ? w2v1 : 0.f };
      v2f A23 = { l3 ? w2v0 : 0.f, l3 ? w2v1 : 0.f };

      // ---- layer 2: K=16 as 4 chained accumulating WMMAs ----
      v8f d2 = __builtin_amdgcn_wmma_f32_16x16x4_f32(false, A20, false, B0,  (short)0, zero8, false, false);
      d2     = __builtin_amdgcn_wmma_f32_16x16x4_f32(false, A21, false, B1c, (short)0, d2,    false, false);
      d2     = __builtin_amdgcn_wmma_f32_16x16x4_f32(false, A22, false, B2c, (short)0, d2,    false, false);
      d2     = __builtin_amdgcn_wmma_f32_16x16x4_f32(false, A23, false, B3c, (short)0, d2,    false, false);

      float h2v[8];
      h2v[0] = elu1(d2[0] + bibA.x); h2v[1] = elu1(d2[1] + bibA.y);
      h2v[2] = elu1(d2[2] + bibA.z); h2v[3] = elu1(d2[3] + bibA.w);
      h2v[4] = elu1(d2[4] + bibB.x); h2v[5] = elu1(d2[5] + bibB.y);
      h2v[6] = elu1(d2[6] + bibB.z); h2v[7] = elu1(d2[7] + bibB.w);

      // ---- layer 3 + softplus (VALU): nets (4g+2h) from vgprs 0-3, (4g+1+2h) from 4-7 ----
      float a02 = fmaf(w02.x, h2v[0], fmaf(w02.y, h2v[1], fmaf(w02.z, h2v[2], fmaf(w02.w, h2v[3], b3A))));
      float a13 = fmaf(w13.x, h2v[4], fmaf(w13.y, h2v[5], fmaf(w13.z, h2v[6], fmaf(w13.w, h2v[7], b3B))));
      float keA = sp1(a02), keB = sp1(a13);

      // scale by gas composition, scatter-add into per-sample channel tile
      float tauA = keA * compW[(m << 3) + gasA];
      float tauB = keB * compW[(m << 3) + gasB];
      atomicAdd(accW + (m << 5) + chA, tauA);
      atomicAdd(accW + (m << 5) + chB, tauB);
    }

    // drain DS pipe (atomics + loads) before readback
    asm volatile("s_wait_dscnt 0x0" ::: "memory");

    // ---- writeback: contiguous 464-float region per output plane ----
    const int base = s0 * 29;
    #pragma unroll
    for (int k = 0; k < 15; ++k) {
      int idx = lane + (k << 5);
      if (idx < 464) {
        int sm = idx / 29;
        int ch = idx - sm * 29;
        out[base + idx] = accW[(sm << 5) + ch];
        float nl = null_lw[s0 + sm];
        float ni = null_iw[s0 + sm];
        float c6 = compW[(sm << 3) + 6];
        float c7 = compW[(sm << 3) + 7];
        outLw[base + idx] = sp1(fmaf(nl, smem[WLW_OFF + ch], smem[BLW_OFF + ch])) * c6;
        outIw[base + idx] = sp1(fmaf(ni, smem[WIW_OFF + ch], smem[BIW_OFF + ch])) * c7;
      }
    }
  }
}

extern "C" void kernel_launch(void* const* d_in, const int* in_sizes, int n_in,
                              void* d_out, int out_size, void* d_ws, size_t ws_size,
                              hipStream_t stream) {
  (void)n_in; (void)out_size; (void)d_ws; (void)ws_size;
  const float* t_p  = (const float*)d_in[0];
  const float* comp = (const float*)d_in[1];
  const float* nlw  = (const float*)d_in[2];
  const float* niw  = (const float*)d_in[3];
  const float* W1   = (const float*)d_in[4];
  const float* b1   = (const float*)d_in[5];
  const float* W2   = (const float*)d_in[6];
  const float* b2   = (const float*)d_in[7];
  const float* W3   = (const float*)d_in[8];
  const float* b3   = (const float*)d_in[9];
  const float* Wlw  = (const float*)d_in[10];
  const float* blw  = (const float*)d_in[11];
  const float* Wiw  = (const float*)d_in[12];
  const float* biw  = (const float*)d_in[13];

  int B = in_sizes[0] / 2;
  int tiles = B >> 4;                 // 16 samples per wave-tile
  int wavesPerBlock = 4;              // blockDim 128 = 4 wave32
  int blocks = (tiles + wavesPerBlock - 1) / wavesPerBlock;
  if (blocks > 1024) blocks = 1024;
  if (blocks < 1) blocks = 1;

  hipLaunchKernelGGL(OpticalDepth_55276229099579_kernel, dim3(blocks), dim3(128), 0, stream,
                     t_p, comp, nlw, niw, W1, b1, W2, b2, W3, b3,
                     Wlw, blw, Wiw, biw, (float*)d_out, B, tiles);
}